// BitNodeTrellis_16303695856318
// MI455X (gfx1250) — compile-verified
//
#include <hip/hip_runtime.h>
#include <cstdint>
#include <cstddef>

// ---------------------------------------------------------------------------
// BitNodeTrellis: out[c,i,j] = logsumexp_k( e1[(u+c)&1, i, k] + e2[c, k, j] )
// per (b1,b2) pair.  Pure streaming kernel -> async global->LDS double-buffered
// pipeline (ASYNCcnt), ds_load_b128 readback, NT b128 stores.
// ---------------------------------------------------------------------------

#define BLOCK            256
#define GRID             1024
#define WAVES_PER_BLOCK  (BLOCK / 32)
#define TILE_PAIRS       32                   // pairs handled per wave-tile
#define TILE_BYTES       (TILE_PAIRS * 32)    // 1024 B per array (e1 or e2)
#define WAVE_LDS_FLOATS  1024                 // 2 buffers * (256 + 256) floats

typedef float v4f __attribute__((ext_vector_type(4)));

__device__ __forceinline__ uint32_t lds_addr_u32(const void* p) {
    // Generic pointers into the LDS aperture carry the workgroup-relative LDS
    // byte offset in their low 32 bits (apertures are 4 GB aligned on gfx12).
    return (uint32_t)(uintptr_t)p;
}

__device__ __forceinline__ void async_copy_b128(uint32_t lds_off, const void* gaddr) {
    // GLOBAL_LOAD_ASYNC_TO_LDS_B128: vdst = LDS byte address, vaddr = 64-bit
    // global address.  Tracked with ASYNCcnt.
    asm volatile("global_load_async_to_lds_b128 %0, %1, off"
                 :
                 : "v"(lds_off), "v"(gaddr)
                 : "memory");
}

__device__ __forceinline__ void prefetch_tile(uint32_t lds_e1, uint32_t lds_e2,
                                              const char* g_e1, const char* g_e2,
                                              int lane) {
    const int o0 = lane * 16;        // lanes 0..31 cover bytes [0, 512)
    const int o1 = 512 + lane * 16;  // lanes 0..31 cover bytes [512, 1024)
    async_copy_b128(lds_e1 + o0, g_e1 + o0);
    async_copy_b128(lds_e1 + o1, g_e1 + o1);
    async_copy_b128(lds_e2 + o0, g_e2 + o0);
    async_copy_b128(lds_e2 + o1, g_e2 + o1);
}

__device__ __forceinline__ float lse2(float x, float y) {
    // Numerically stable 2-way logsumexp using HW v_exp_f32 / v_log_f32.
    float m = fmaxf(x, y);
    float d = fminf(x, y) - m;          // <= 0
    return m + __logf(1.0f + __expf(d));
}

__global__ __launch_bounds__(BLOCK)
void BitNodeTrellis_kernel(const float* __restrict__ e1,
                           const float* __restrict__ e2,
                           const int*   __restrict__ uhat,
                           float*       __restrict__ out,
                           int numPairs) {
    __shared__ alignas(16) float lds_f[WAVES_PER_BLOCK * WAVE_LDS_FLOATS];

    const int lane        = threadIdx.x & 31;
    const int waveInBlock = threadIdx.x >> 5;
    const int waveId      = blockIdx.x * WAVES_PER_BLOCK + waveInBlock;
    const int totalWaves  = GRID * WAVES_PER_BLOCK;
    const int numTiles    = numPairs / TILE_PAIRS;   // 65536 for this problem

    // Wave-private LDS region: buffer b -> e1 @ [b*512, b*512+256) floats,
    //                                      e2 @ [b*512+256, b*512+512) floats.
    float* wbase = &lds_f[waveInBlock * WAVE_LDS_FLOATS];
    uint32_t l_e1[2], l_e2[2];
    l_e1[0] = lds_addr_u32(wbase + 0);
    l_e2[0] = lds_addr_u32(wbase + 256);
    l_e1[1] = lds_addr_u32(wbase + 512);
    l_e2[1] = lds_addr_u32(wbase + 768);

    const char* e1c = (const char*)e1;
    const char* e2c = (const char*)e2;

    // Prime the pipeline with this wave's first tile.
    if (waveId < numTiles) {
        prefetch_tile(l_e1[0], l_e2[0],
                      e1c + (size_t)waveId * TILE_BYTES,
                      e2c + (size_t)waveId * TILE_BYTES, lane);
    }

    int buf = 0;
    for (int t = waveId; t < numTiles; t += totalWaves) {
        const int  tn      = t + totalWaves;
        const bool hasNext = tn < numTiles;       // wave-uniform

        if (hasNext) {
            prefetch_tile(l_e1[buf ^ 1], l_e2[buf ^ 1],
                          e1c + (size_t)tn * TILE_BYTES,
                          e2c + (size_t)tn * TILE_BYTES, lane);
            // 4 ops for next tile in flight; wait until current tile's 4 done.
            asm volatile("s_wait_asynccnt 4" ::: "memory");
        } else {
            asm volatile("s_wait_asynccnt 0" ::: "memory");
        }

        const size_t pair = (size_t)t * TILE_PAIRS + lane;
        const int u = uhat[pair] & 1;

        // This lane's 2x2x2 blocks from LDS (ds_load_b128 x4).
        const v4f* Ap = (const v4f*)(wbase + buf * 512 + lane * 8);
        const v4f* Bp = (const v4f*)(wbase + buf * 512 + 256 + lane * 8);
        v4f aLo = Ap[0];   // e1 sel=0: [i0k0, i0k1, i1k0, i1k1]
        v4f aHi = Ap[1];   // e1 sel=1
        v4f bLo = Bp[0];   // e2 c=0 : [k0j0, k0j1, k1j0, k1j1]
        v4f bHi = Bp[1];   // e2 c=1

        // c=0 uses e1 half (u), c=1 uses e1 half (u^1).
        v4f s0 = u ? aHi : aLo;
        v4f s1 = u ? aLo : aHi;

        v4f o0, o1;
        o0.x = lse2(s0.x + bLo.x, s0.y + bLo.z);
        o0.y = lse2(s0.x + bLo.y, s0.y + bLo.w);
        o0.z = lse2(s0.z + bLo.x, s0.w + bLo.z);
        o0.w = lse2(s0.z + bLo.y, s0.w + bLo.w);
        o1.x = lse2(s1.x + bHi.x, s1.y + bHi.z);
        o1.y = lse2(s1.x + bHi.y, s1.y + bHi.w);
        o1.z = lse2(s1.z + bHi.x, s1.w + bHi.z);
        o1.w = lse2(s1.z + bHi.y, s1.w + bHi.w);

        // Write-once output: non-temporal b128 stores, fully coalesced.
        v4f* outv = (v4f*)(out + pair * 8);
        __builtin_nontemporal_store(o0, outv);
        __builtin_nontemporal_store(o1, outv + 1);

        buf ^= 1;
    }
}

extern "C" void kernel_launch(void* const* d_in, const int* in_sizes, int n_in,
                              void* d_out, int out_size, void* d_ws, size_t ws_size,
                              hipStream_t stream) {
    const float* e1   = (const float*)d_in[0];
    const float* e2   = (const float*)d_in[1];
    const int*   uhat = (const int*)d_in[2];
    float*       out  = (float*)d_out;
    const int numPairs = in_sizes[2];   // B1*B2 = 2,097,152 (divisible by 32)

    BitNodeTrellis_kernel<<<GRID, BLOCK, 0, stream>>>(e1, e2, uhat, out, numPairs);
}